// BatchelorGPUNUFFTFwd_64458869179026
// MI455X (gfx1250) — compile-verified
//
#include <hip/hip_runtime.h>
#include <hip/hip_bf16.h>
#include <math.h>

// BatchelorGPUNUFFT forward, factorized NDFT on CDNA5 (gfx1250, wave32).
//
// Stage 0 (warp_coil_kernel): gT[t][c][j][i] = csm[c,i,j] * x[warp(i,j,t)]
// Stage 1+2 (nufft_wmma_kernel): per m-tile of 16 k-space points,
//   inner[m,(c,i)] = sum_j Ey[m,j] * gT[.,j,i]   via V_WMMA_F32_16X16X4_F32
//   out[m,c]      += sum_i Ex[m,i] * inner[m,(c,i)]  (VALU + cross-lane reduce)
// fp32 WMMA keeps full reference precision (oscillatory 16K-term sums).

typedef __attribute__((ext_vector_type(2))) float v2f;
typedef __attribute__((ext_vector_type(8))) float v8f;

#define NR   128   // N_READ
#define NS   16    // N_SPOKES
#define NC   4     // N_COILS
#define NT   4     // N_T
#define TWO_PI 6.28318530717958647692f

// ---------------------------------------------------------------------------
// Kernel 1: warp + coil weighting, stored j-major for coalesced B-tile loads.
// gT[((t*NC + c)*NR + j)*NR + i] = csm[c,i,j] * x[si,sj]
// ---------------------------------------------------------------------------
__global__ __launch_bounds__(256) void warp_coil_kernel(
    const float* __restrict__ x, const float* __restrict__ csm,
    const float* __restrict__ flow, float* __restrict__ gT) {
  int idx = blockIdx.x * blockDim.x + threadIdx.x;  // 0..16383
  int i = idx >> 7;
  int j = idx & 127;
#pragma unroll
  for (int t = 0; t < NT; ++t) {
    // flow shape [NR, NR, 2, NT]
    float f0 = flow[((i * NR + j) * 2 + 0) * NT + t];  // row displacement
    float f1 = flow[((i * NR + j) * 2 + 1) * NT + t];  // col displacement
    int si = (int)rintf((float)i + f0); si = si < 0 ? 0 : (si > 127 ? 127 : si);
    int sj = (int)rintf((float)j + f1); sj = sj < 0 ? 0 : (sj > 127 ? 127 : sj);
    float v = x[si * NR + sj];
#pragma unroll
    for (int c = 0; c < NC; ++c) {
      gT[((t * NC + c) * NR + j) * NR + i] = csm[(c * NR + i) * NR + j] * v;
    }
  }
}

// ---------------------------------------------------------------------------
// Kernel 2: WMMA NDFT. grid.x = 2048/16 = 128 m-tiles; 256 threads = 8 waves.
// Wave w: coil c = w&3, i-half ihalf = w>>2 (i in [ihalf*64, ihalf*64+64)).
// ---------------------------------------------------------------------------
__global__ __launch_bounds__(256) void nufft_wmma_kernel(
    const float* __restrict__ traj, const float* __restrict__ gT,
    float* __restrict__ out) {
  __shared__ float smem[2][16][NC][2];

  const int tid   = threadIdx.x;
  const int w     = tid >> 5;
  const int lane  = tid & 31;
  const int c     = w & 3;
  const int ihalf = w >> 2;
  const int n     = lane & 15;   // A-row index AND B/D column index
  const int h     = lane >> 4;   // lane half (K interleave)

  const int m0    = blockIdx.x * 16;  // global m = s*NR + rr, m-tile stays in one spoke
  const int s     = m0 >> 7;
  const int rr0   = m0 & 127;
  const int m_row = m0 + n;           // this lane's A-matrix row

  float acc2_r[8], acc2_i[8];
#pragma unroll
  for (int v = 0; v < 8; ++v) { acc2_r[v] = 0.f; acc2_i[v] = 0.f; }

  for (int t = 0; t < NT; ++t) {
    // traj shape [NS, NR, NT, 2]; flat = ((m*NT)+t)*2 + d
    const float kx = traj[(m_row * NT + t) * 2 + 0];  // multiplies (i-64)
    const float ky = traj[(m_row * NT + t) * 2 + 1];  // multiplies (j-64)

    v8f Dr[4], Di[4];
#pragma unroll
    for (int blk = 0; blk < 4; ++blk) {
#pragma unroll
      for (int q = 0; q < 8; ++q) { Dr[blk][q] = 0.f; Di[blk][q] = 0.f; }
    }

    const float* gbase = gT + (t * NC + c) * NR * NR;

    // K loop over j in steps of 4 (WMMA 16x16x4 fp32)
    for (int j0 = 0; j0 < NR; j0 += 4) {
      // A tiles (Ey cos / sin). Layout: lane l holds M=l%16; VGPR p -> K = p + 2h.
      const int jA = j0 + 2 * h;
      float s0, c0, s1, c1;
      __sincosf(-TWO_PI * ky * (float)(jA + 0 - 64), &s0, &c0);
      __sincosf(-TWO_PI * ky * (float)(jA + 1 - 64), &s1, &c1);
      v2f Ar = {c0, c1};  // real part of Ey
      v2f Ai = {s0, s1};  // imag part of Ey

      // B tiles: B[k,n] = gT[.., j0+k, i0+n]; VGPR p -> K = p + 2h (mirrors A).
      const float* gb = gbase + jA * NR + ihalf * 64 + n;
#pragma unroll
      for (int blk = 0; blk < 4; ++blk) {
        v2f B = { gb[blk * 16], gb[blk * 16 + NR] };  // rows jA, jA+1
        Dr[blk] = __builtin_amdgcn_wmma_f32_16x16x4_f32(
            false, Ar, false, B, (short)0, Dr[blk], false, false);
        Di[blk] = __builtin_amdgcn_wmma_f32_16x16x4_f32(
            false, Ai, false, B, (short)0, Di[blk], false, false);
      }
    }

    // Stage 2: apply Ex[m,i] to inner and accumulate per-lane.
    // D layout: lane l, VGPR v -> (m_local = v + 8h, i_col = l%16).
#pragma unroll
    for (int blk = 0; blk < 4; ++blk) {
      const float xi = (float)(ihalf * 64 + blk * 16 + n - 64);
#pragma unroll
      for (int v = 0; v < 8; ++v) {
        float kxv = __shfl(kx, v + 8 * h, 32);  // kx for m_local = v + 8h
        float exr, exi;
        __sincosf(-TWO_PI * kxv * xi, &exi, &exr);
        float dr = Dr[blk][v], di = Di[blk][v];
        acc2_r[v] += exr * dr - exi * di;
        acc2_i[v] += exr * di + exi * dr;
      }
    }
  }

  // Reduce over i within each 16-lane half (m differs between halves).
#pragma unroll
  for (int v = 0; v < 8; ++v) {
#pragma unroll
    for (int mask = 1; mask < 16; mask <<= 1) {
      acc2_r[v] += __shfl_xor(acc2_r[v], mask, 32);
      acc2_i[v] += __shfl_xor(acc2_i[v], mask, 32);
    }
  }

  if (n == 0) {  // lanes 0 (h=0, m=v) and 16 (h=1, m=v+8)
#pragma unroll
    for (int v = 0; v < 8; ++v) {
      smem[ihalf][v + 8 * h][c][0] = acc2_r[v];
      smem[ihalf][v + 8 * h][c][1] = acc2_i[v];
    }
  }
  __syncthreads();

  // Combine the two i-halves; each block writes its 16 m x 4 c x 2 floats once.
  if (tid < 128) {
    int m_local = tid >> 3;
    int c2      = (tid >> 1) & 3;
    int comp    = tid & 1;
    float val = smem[0][m_local][c2][comp] + smem[1][m_local][c2][comp];
    int rr = rr0 + m_local;
    // out shape [1, NR, NS, NC] complex64 -> interleaved float32
    out[((rr * NS + s) * NC + c2) * 2 + comp] = val;
  }
}

// ---------------------------------------------------------------------------
extern "C" void kernel_launch(void* const* d_in, const int* in_sizes, int n_in,
                              void* d_out, int out_size, void* d_ws, size_t ws_size,
                              hipStream_t stream) {
  const float* x    = (const float*)d_in[0];  // [128,128]
  const float* traj = (const float*)d_in[1];  // [16,128,4,2]
  const float* csm  = (const float*)d_in[2];  // [4,128,128]
  // d_in[3] = dcf: unused by the reference
  const float* flow = (const float*)d_in[4];  // [128,128,2,4]

  float* gT  = (float*)d_ws;   // NT*NC*128*128 floats = 1 MB scratch
  float* out = (float*)d_out;  // 128*16*4 complex64 = 16384 floats

  warp_coil_kernel<<<dim3(64), dim3(256), 0, stream>>>(x, csm, flow, gT);
  nufft_wmma_kernel<<<dim3(128), dim3(256), 0, stream>>>(traj, gT, out);
}